// MultiHeadAttention_71073118814661
// MI455X (gfx1250) — compile-verified
//
#include <hip/hip_runtime.h>

#define BB  16
#define SS  512
#define HH  32
#define HIDN 512
#define DKK 16

typedef __attribute__((ext_vector_type(2))) float v2f;
typedef __attribute__((ext_vector_type(8))) float v8f;

__device__ __forceinline__ v2f ld2(const float* p) { return *(const v2f*)p; }

// fp32 WMMA: D(16x16,f32) = A(16x4,f32) x B(4x16,f32) + C
__device__ __forceinline__ v8f wmma4(v2f a, v2f b, v8f c) {
    return __builtin_amdgcn_wmma_f32_16x16x4_f32(false, a, false, b, (short)0, c,
                                                 false, false);
}

// ---- CDNA5 async global->LDS path (guarded; falls back to direct loads) ----
#if defined(__has_builtin)
#  if __has_builtin(__builtin_amdgcn_global_load_async_to_lds_b128) && \
      __has_builtin(__builtin_amdgcn_s_wait_asynccnt)
#    define USE_ASYNC_LDS 1
#  endif
#endif
#ifndef USE_ASYNC_LDS
#  define USE_ASYNC_LDS 0
#endif

#if USE_ASYNC_LDS
// Builtin signature (from hipcc diagnostic): first param is
// 'int __vector_size__(16) __device__ *', i.e. addrspace(1) v4i*.
typedef __attribute__((__vector_size__(4 * sizeof(int)))) int v4i;
typedef __attribute__((address_space(1))) v4i gv4i;   // global (prints __device__)
typedef __attribute__((address_space(3))) v4i lv4i;   // LDS

// One lane moves 16B (4 floats) of a 16x16 f32 bias tile: id in [0,64).
__device__ __forceinline__ void async_bias_chunk(const float* gbase, int k0,
                                                 int id, float* lbase) {
    const int row = id >> 2;        // 16 rows
    const int ch  = id & 3;         // 4 x 16B chunks per 64B row
    const float* g = gbase + (size_t)row * SS + k0 + ch * 4;
    float*       l = lbase + row * 16 + ch * 4;
    __builtin_amdgcn_global_load_async_to_lds_b128(
        (gv4i*)(unsigned long long)g,
        (lv4i*)(unsigned)(unsigned long long)l, 0, 0);
}
#endif

// ---------------------------------------------------------------------------
// Kernel 1: fused QKV projection.
// out[b,h,s,d] = (X[b,s,:] . W[h*16+d,:] + bias[h*16+d]) * (proj==0 ? 1/4 : 1)
// ---------------------------------------------------------------------------
__global__ void qkv_proj_kernel(const float* __restrict__ q,
                                const float* __restrict__ k,
                                const float* __restrict__ v,
                                const float* __restrict__ Wq, const float* __restrict__ bq,
                                const float* __restrict__ Wk, const float* __restrict__ bk,
                                const float* __restrict__ Wv, const float* __restrict__ bv,
                                float* __restrict__ qh, float* __restrict__ kh,
                                float* __restrict__ vh) {
    const int lane = threadIdx.x;         // 0..31 (wave32)
    const int half = lane >> 4;
    const int l16  = lane & 15;
    const int n0   = blockIdx.x * 16;
    const int m0   = blockIdx.y * 16;
    const int proj = blockIdx.z;

    const float *X, *W, *bias; float* out; float scale;
    if (proj == 0)      { X = q; W = Wq; bias = bq; out = qh; scale = 0.25f; } // DK^-0.5
    else if (proj == 1) { X = k; W = Wk; bias = bk; out = kh; scale = 1.0f; }
    else                { X = v; W = Wv; bias = bv; out = vh; scale = 1.0f; }

    const float* aRow = X + (size_t)(m0 + l16) * HIDN + half * 2;
    const float* bRow = W + (size_t)(n0 + l16) * HIDN + half * 2;

    v8f acc = {};
#pragma unroll 8
    for (int kk = 0; kk < HIDN; kk += 4)
        acc = wmma4(ld2(aRow + kk), ld2(bRow + kk), acc);

    const int bidx = m0 >> 9;
    const int s0   = m0 & (SS - 1);
    const int h    = n0 >> 4;
    const float bsc = bias[n0 + l16];
    float* dst = out + ((size_t)(bidx * HH + h) * SS + s0) * DKK;
#pragma unroll
    for (int r = 0; r < 8; ++r) {
        int row = r + half * 8;           // C layout: vgpr r -> rows r / r+8
        dst[row * DKK + l16] = (acc[r] + bsc) * scale;
    }
}

// ---------------------------------------------------------------------------
// Kernel 2: flash-style attention for one (b, h, 16-row q tile).
// Single pass over the 0.54 GB attn_bias stream; online softmax kept in
// C-layout VGPRs; 1 KB LDS p-tile for C->A relayout; bias tile double-
// buffered into LDS with GLOBAL_LOAD_ASYNC_TO_LDS_B128 when available.
// ---------------------------------------------------------------------------
__global__ void attention_kernel(const float* __restrict__ qh,
                                 const float* __restrict__ kh,
                                 const float* __restrict__ vh,
                                 const float* __restrict__ attn_bias,
                                 const float* __restrict__ rel_pos,
                                 float* __restrict__ attn_out) {
    __shared__ __align__(16) float pbuf[16 * 16];       // 1 KB p-tile
#if USE_ASYNC_LDS
    __shared__ __align__(16) float bbuf[2][16][16];     // 2 KB bias double buffer
#endif

    const int lane = threadIdx.x;
    const int half = lane >> 4;
    const int l16  = lane & 15;
    const int qt = blockIdx.x;
    const int h  = blockIdx.y;
    const int b  = blockIdx.z;
    const int bh = b * HH + h;
    const int s0 = qt * 16;

    // Preload A (q tile): lane = row l16, K = 4t + half*2 + {0,1}
    const float* qp = qh + ((size_t)bh * SS + s0 + l16) * DKK + half * 2;
    const v2f aQ0 = ld2(qp + 0);
    const v2f aQ1 = ld2(qp + 4);
    const v2f aQ2 = ld2(qp + 8);
    const v2f aQ3 = ld2(qp + 12);

    const float* biasBase = attn_bias + ((size_t)bh * SS + s0) * SS;
    const float* rpBase   = rel_pos + (size_t)b * (SS - 1) * (SS - 1);

    float m[8], l[8];
#pragma unroll
    for (int r = 0; r < 8; ++r) { m[r] = -3.402823e38f; l[r] = 0.0f; }
    v8f o = {};

#if USE_ASYNC_LDS
    // preload bias tile 0 (2 x b128 per lane covers 16x16 f32 = 1 KB)
    async_bias_chunk(biasBase, 0, lane,      &bbuf[0][0][0]);
    async_bias_chunk(biasBase, 0, lane + 32, &bbuf[0][0][0]);
#endif

    for (int j = 0; j < SS / 16; ++j) {
        const int k0 = j * 16;

        // scores tile: B = kh^T, lane holds col l16
        const float* kp = kh + ((size_t)bh * SS + k0 + l16) * DKK + half * 2;
        v8f acc = {};
        acc = wmma4(aQ0, ld2(kp + 0),  acc);
        acc = wmma4(aQ1, ld2(kp + 4),  acc);
        acc = wmma4(aQ2, ld2(kp + 8),  acc);
        acc = wmma4(aQ3, ld2(kp + 12), acc);

#if USE_ASYNC_LDS
        if (j + 1 < SS / 16) {
            async_bias_chunk(biasBase, k0 + 16, lane,      &bbuf[(j + 1) & 1][0][0]);
            async_bias_chunk(biasBase, k0 + 16, lane + 32, &bbuf[(j + 1) & 1][0][0]);
            __builtin_amdgcn_s_wait_asynccnt(2);   // tile j resident (in-order)
        } else {
            __builtin_amdgcn_s_wait_asynccnt(0);
        }
#else
        if (j + 1 < SS / 16)
            __builtin_prefetch(biasBase + (half * 8) * SS + k0 + 16 + l16, 0, 0);
#endif

        // bias + distance mask + online softmax update (C layout)
#pragma unroll
        for (int r = 0; r < 8; ++r) {
            const int row = r + half * 8;
            const int qg  = s0 + row;
            const int kg  = k0 + l16;
#if USE_ASYNC_LDS
            float bia = bbuf[j & 1][row][l16];
#else
            float bia = biasBase[(size_t)row * SS + kg];
#endif
            if (qg >= 1 && kg >= 1) {
                float rp  = rpBase[(size_t)(qg - 1) * (SS - 1) + (kg - 1)];
                bool flag = (rp <= 5.5f) || (bia < -1e8f);
                bia = flag ? bia : 0.0f;
            }
            float s = acc[r] + bia;

            // row max over the 16-lane column group (wave32 shuffles)
            float tm = s;
            tm = fmaxf(tm, __shfl_xor(tm, 1, 32));
            tm = fmaxf(tm, __shfl_xor(tm, 2, 32));
            tm = fmaxf(tm, __shfl_xor(tm, 4, 32));
            tm = fmaxf(tm, __shfl_xor(tm, 8, 32));
            float nm = fmaxf(m[r], tm);
            float sc = __expf(m[r] - nm);
            float pe = __expf(s - nm);
            float ps = pe;
            ps += __shfl_xor(ps, 1, 32);
            ps += __shfl_xor(ps, 2, 32);
            ps += __shfl_xor(ps, 4, 32);
            ps += __shfl_xor(ps, 8, 32);
            l[r] = l[r] * sc + ps;
            m[r] = nm;
            o[r] *= sc;
            pbuf[row * 16 + l16] = pe;      // C layout -> LDS for A relayout
        }
        __syncthreads();                    // single wave: waitcnt only

        // o += p @ V_tile   (A read back from LDS in A layout)
#pragma unroll
        for (int t = 0; t < 4; ++t) {
            const int kd = k0 + 4 * t + half * 2;
            v2f a = ld2(&pbuf[l16 * 16 + 4 * t + half * 2]);
            v2f bm;
            bm.x = vh[((size_t)bh * SS + kd)     * DKK + l16];
            bm.y = vh[((size_t)bh * SS + kd + 1) * DKK + l16];
            o = wmma4(a, bm, o);
        }
        __syncthreads();
    }

    // epilogue: divide by running sum, store to [B,S,HID] layout
    float* dst = attn_out + ((size_t)b * SS + s0) * HIDN + h * DKK;
#pragma unroll
    for (int r = 0; r < 8; ++r) {
        int row = r + half * 8;
        dst[(size_t)row * HIDN + l16] = o[r] / l[r];
    }
}

// ---------------------------------------------------------------------------
// Kernel 3: output projection  out = X @ Wo^T + bo
// ---------------------------------------------------------------------------
__global__ void out_proj_kernel(const float* __restrict__ x,
                                const float* __restrict__ Wo,
                                const float* __restrict__ bo,
                                float* __restrict__ out) {
    const int lane = threadIdx.x;
    const int half = lane >> 4;
    const int l16  = lane & 15;
    const int n0 = blockIdx.x * 16;
    const int m0 = blockIdx.y * 16;

    const float* aRow = x  + (size_t)(m0 + l16) * HIDN + half * 2;
    const float* bRow = Wo + (size_t)(n0 + l16) * HIDN + half * 2;

    v8f acc = {};
#pragma unroll 8
    for (int kk = 0; kk < HIDN; kk += 4)
        acc = wmma4(ld2(aRow + kk), ld2(bRow + kk), acc);

    const float bb = bo[n0 + l16];
    float* dst = out + (size_t)m0 * HIDN + n0;
#pragma unroll
    for (int r = 0; r < 8; ++r) {
        int row = r + half * 8;
        dst[(size_t)row * HIDN + l16] = acc[r] + bb;
    }
}

extern "C" void kernel_launch(void* const* d_in, const int* in_sizes, int n_in,
                              void* d_out, int out_size, void* d_ws, size_t ws_size,
                              hipStream_t stream) {
    (void)in_sizes; (void)n_in; (void)out_size; (void)ws_size;
    const float* q         = (const float*)d_in[0];
    const float* k         = (const float*)d_in[1];
    const float* v         = (const float*)d_in[2];
    const float* attn_bias = (const float*)d_in[3];
    const float* rel_pos   = (const float*)d_in[4];
    const float* Wq = (const float*)d_in[5];
    const float* bq = (const float*)d_in[6];
    const float* Wk = (const float*)d_in[7];
    const float* bk = (const float*)d_in[8];
    const float* Wv = (const float*)d_in[9];
    const float* bv = (const float*)d_in[10];
    const float* Wo = (const float*)d_in[11];
    const float* bo = (const float*)d_in[12];

    // Workspace: qh | kh | vh | attn_out, each B*H*S*DK = 4M floats (64 MB total)
    const size_t SZ = (size_t)BB * HH * SS * DKK;
    float* qh = (float*)d_ws;
    float* kh = qh + SZ;
    float* vh = kh + SZ;
    float* ao = vh + SZ;

    qkv_proj_kernel<<<dim3(HIDN / 16, BB * SS / 16, 3), 32, 0, stream>>>(
        q, k, v, Wq, bq, Wk, bk, Wv, bv, qh, kh, vh);

    attention_kernel<<<dim3(SS / 16, HH, BB), 32, 0, stream>>>(
        qh, kh, vh, attn_bias, rel_pos, ao);

    out_proj_kernel<<<dim3(HIDN / 16, BB * SS / 16), 32, 0, stream>>>(
        ao, Wo, bo, (float*)d_out);
}